// discriminative_loss_70952859730216
// MI455X (gfx1250) — compile-verified
//
#include <hip/hip_runtime.h>
#include <hip/hip_bf16.h>

// ---------------------------------------------------------------------------
// Discriminative loss for MI455X (gfx1250).
//   B=8, C=128, N=65536, L=64
//   Pass 1: streaming 256MB read of prediction; segmented per-label sums via
//           LDS ds_add_f32 (stride-129 rows -> conflict-free across labels),
//           plus per-point channel-mean variance term.
//   Pass 2: per-batch finalize; Gram = centers^T * centers over L=64 using
//           V_WMMA_F32_16X16X4_F32 (exact f32), hinge/dist/var/reg reduction.
// ---------------------------------------------------------------------------

#define BB   8
#define CC   128
#define NN   65536
#define LBL  64

#define BLOCKS_PER_BATCH 32
#define PTS_PER_BLOCK    2048   // 65536 / 32
#define T1               256    // threads, pass 1

// per-batch workspace layout (floats): centers_sum[64*128] | counts[64] | vsum[64]
#define WS_CENT   0
#define WS_CNT    (LBL * CC)          // 8192
#define WS_VS     (WS_CNT + LBL)      // 8256
#define WS_STRIDE (WS_VS + LBL)       // 8320
#define WS_TOTAL  (BB * WS_STRIDE)    // 66560 floats

#define D_VAR  0.5f
#define D_DIST 1.5f
#define P_VAR  1.0f
#define P_DIST 1.0f
#define P_REG  0.001f

typedef __attribute__((ext_vector_type(2))) float v2f;
typedef __attribute__((ext_vector_type(8))) float v8f;

// ---------------------------------------------------------------------------
// Kernel 0: zero workspace accumulators and the output scalar.
// ---------------------------------------------------------------------------
__global__ void dl_zero_kernel(float* __restrict__ ws, float* __restrict__ out,
                               int nws, int nout) {
  int i = blockIdx.x * blockDim.x + threadIdx.x;
  if (i < nws)  ws[i]  = 0.0f;
  if (i < nout) out[i] = 0.0f;
}

// ---------------------------------------------------------------------------
// Kernel 1: streaming pass over prediction [B,C,N].
//   grid = B * BLOCKS_PER_BATCH blocks of 256 threads.
//   Each thread owns 4 consecutive points (float4 loads, coalesced 512B/wave).
// ---------------------------------------------------------------------------
__global__ __launch_bounds__(T1) void dl_pass1_kernel(
    const float* __restrict__ pred, const int* __restrict__ label,
    float* __restrict__ ws) {
  // stride-129 rows: LDS bank = (129*l + c) % 64 = (l + c) % 64 -> distinct
  // labels never collide on a bank; only same-label atomics serialize.
  __shared__ float    scent[LBL * 129];
  __shared__ unsigned scnt[LBL];
  __shared__ float    svs[LBL];

  const int tid   = threadIdx.x;
  const int b     = blockIdx.x >> 5;     // / BLOCKS_PER_BATCH
  const int chunk = blockIdx.x & 31;

  for (int i = tid; i < LBL * 129; i += T1) scent[i] = 0.0f;
  if (tid < LBL) { scnt[tid] = 0u; svs[tid] = 0.0f; }
  __syncthreads();

  const float* pb = pred + (size_t)b * CC * NN;
  const int*   lb = label + (size_t)b * NN;

  for (int it = 0; it < PTS_PER_BLOCK / (T1 * 4); ++it) {
    const int n4 = chunk * PTS_PER_BLOCK + it * (T1 * 4) + tid * 4;
    const int4 lab = *(const int4*)(lb + n4);
    const int a0 = lab.x * 129, a1 = lab.y * 129;
    const int a2 = lab.z * 129, a3 = lab.w * 129;

    float s0 = 0.f, s1 = 0.f, s2 = 0.f, s3 = 0.f;
    float q0 = 0.f, q1 = 0.f, q2 = 0.f, q3 = 0.f;

#pragma unroll 4
    for (int c = 0; c < CC; ++c) {
      const float4 x = *(const float4*)(pb + (size_t)c * NN + n4);
      s0 += x.x; q0 = fmaf(x.x, x.x, q0);
      s1 += x.y; q1 = fmaf(x.y, x.y, q1);
      s2 += x.z; q2 = fmaf(x.z, x.z, q2);
      s3 += x.w; q3 = fmaf(x.w, x.w, q3);
      __hip_atomic_fetch_add(&scent[a0 + c], x.x, __ATOMIC_RELAXED, __HIP_MEMORY_SCOPE_WORKGROUP);
      __hip_atomic_fetch_add(&scent[a1 + c], x.y, __ATOMIC_RELAXED, __HIP_MEMORY_SCOPE_WORKGROUP);
      __hip_atomic_fetch_add(&scent[a2 + c], x.z, __ATOMIC_RELAXED, __HIP_MEMORY_SCOPE_WORKGROUP);
      __hip_atomic_fetch_add(&scent[a3 + c], x.w, __ATOMIC_RELAXED, __HIP_MEMORY_SCOPE_WORKGROUP);
    }

    // v = relu(||x - mean_c(x)|| - D_VAR)^2, ||.||^2 = sumsq - sum^2/C
    const float inv_c = 1.0f / (float)CC;
    float pn, v;
    pn = sqrtf(fmaxf(q0 - s0 * s0 * inv_c, 0.f)); v = fmaxf(pn - D_VAR, 0.f); v *= v;
    __hip_atomic_fetch_add(&svs[lab.x], v, __ATOMIC_RELAXED, __HIP_MEMORY_SCOPE_WORKGROUP);
    pn = sqrtf(fmaxf(q1 - s1 * s1 * inv_c, 0.f)); v = fmaxf(pn - D_VAR, 0.f); v *= v;
    __hip_atomic_fetch_add(&svs[lab.y], v, __ATOMIC_RELAXED, __HIP_MEMORY_SCOPE_WORKGROUP);
    pn = sqrtf(fmaxf(q2 - s2 * s2 * inv_c, 0.f)); v = fmaxf(pn - D_VAR, 0.f); v *= v;
    __hip_atomic_fetch_add(&svs[lab.z], v, __ATOMIC_RELAXED, __HIP_MEMORY_SCOPE_WORKGROUP);
    pn = sqrtf(fmaxf(q3 - s3 * s3 * inv_c, 0.f)); v = fmaxf(pn - D_VAR, 0.f); v *= v;
    __hip_atomic_fetch_add(&svs[lab.w], v, __ATOMIC_RELAXED, __HIP_MEMORY_SCOPE_WORKGROUP);

    __hip_atomic_fetch_add(&scnt[lab.x], 1u, __ATOMIC_RELAXED, __HIP_MEMORY_SCOPE_WORKGROUP);
    __hip_atomic_fetch_add(&scnt[lab.y], 1u, __ATOMIC_RELAXED, __HIP_MEMORY_SCOPE_WORKGROUP);
    __hip_atomic_fetch_add(&scnt[lab.z], 1u, __ATOMIC_RELAXED, __HIP_MEMORY_SCOPE_WORKGROUP);
    __hip_atomic_fetch_add(&scnt[lab.w], 1u, __ATOMIC_RELAXED, __HIP_MEMORY_SCOPE_WORKGROUP);
  }
  __syncthreads();

  // flush block-local accumulators to global workspace (L2-resident, 266KB)
  float* wb = ws + (size_t)b * WS_STRIDE;
  for (int i = tid; i < LBL * CC; i += T1) {
    const float val = scent[(i >> 7) * 129 + (i & 127)];
    __hip_atomic_fetch_add(&wb[WS_CENT + i], val, __ATOMIC_RELAXED, __HIP_MEMORY_SCOPE_AGENT);
  }
  if (tid < LBL) {
    __hip_atomic_fetch_add(&wb[WS_CNT + tid], (float)scnt[tid], __ATOMIC_RELAXED, __HIP_MEMORY_SCOPE_AGENT);
    __hip_atomic_fetch_add(&wb[WS_VS + tid], svs[tid], __ATOMIC_RELAXED, __HIP_MEMORY_SCOPE_AGENT);
  }
}

// ---------------------------------------------------------------------------
// Kernel 2: finalize. grid = 8 blocks (one per batch) of 128 threads (4 waves).
//   Gram[c,d] = sum_l centers[l,c]*centers[l,d] via V_WMMA_F32_16X16X4_F32.
// ---------------------------------------------------------------------------
__global__ __launch_bounds__(128) void dl_finalize_kernel(
    const float* __restrict__ ws, float* __restrict__ out) {
  // stride 144: bank = (16*l + c) % 64; for the WMMA A/B reads the two lane
  // half-groups hit rows k and k+2 (banks offset by 32) -> conflict-free.
  __shared__ float cent[LBL * 144];
  __shared__ float sq[CC];
  __shared__ float cnts[LBL];
  __shared__ float red[128];

  const int tid = threadIdx.x;
  const int b   = blockIdx.x;
  const float* wb = ws + (size_t)b * WS_STRIDE;

  if (tid < LBL) cnts[tid] = wb[WS_CNT + tid];
  __syncthreads();

  // centers = sums / counts
  for (int i = tid; i < LBL * CC; i += 128) {
    const int l = i >> 7, c = i & 127;
    cent[l * 144 + c] = wb[WS_CENT + i] / cnts[l];
  }
  __syncthreads();

  // column norms sqn[c] = sum_l centers[l,c]^2 ; l_reg term = sqrt(sqn[c])
  float lreg_c;
  {
    const int c = tid;
    float s = 0.f;
    for (int l = 0; l < LBL; ++l) {
      const float x = cent[l * 144 + c];
      s = fmaf(x, x, s);
    }
    sq[c] = s;
    lreg_c = sqrtf(s);
  }
  // var term: threads 0..63 own one label
  const float vterm = (tid < LBL) ? (wb[WS_VS + tid] / cnts[tid]) : 0.0f;
  __syncthreads();

  // ---- Gram via f32 WMMA: 8x8 tiles of 16x16, K = 64 in steps of 4 ----
  const int wv   = tid >> 5;   // wave id 0..3 (uniform per wave)
  const int lane = tid & 31;
  const int m    = lane & 15;
  const int kh   = lane >> 4;  // lane half-group selects K {0,1} vs {2,3} rows

  float hinge = 0.0f;
  for (int t = wv; t < 64; t += 4) {     // 16 tiles per wave, uniform indices
    const int ci = t >> 3;               // row tile (c block)
    const int cj = t & 7;                // col tile (d block)
    const int arow = ci * 16 + m;
    const int brow = cj * 16 + m;

    v8f acc = {0.f, 0.f, 0.f, 0.f, 0.f, 0.f, 0.f, 0.f};
    for (int k0 = 0; k0 < LBL; k0 += 4) {
      const int ka = k0 + 2 * kh;
      v2f a, bmat;
      // A(16x4): A[m][k] = centers[k][ci*16+m]; VGPR0 holds K=0/2, VGPR1 K=1/3
      a.x    = cent[ka * 144 + arow];
      a.y    = cent[(ka + 1) * 144 + arow];
      // B(4x16): B[k][n] = centers[k][cj*16+n]; same lane pattern
      bmat.x = cent[ka * 144 + brow];
      bmat.y = cent[(ka + 1) * 144 + brow];
      acc = __builtin_amdgcn_wmma_f32_16x16x4_f32(
          /*neg_a=*/false, a, /*neg_b=*/false, bmat,
          /*c_mod=*/(short)0, acc, /*reuse_a=*/false, /*reuse_b=*/false);
    }

    // D layout: VGPR r -> row M = r + 8*kh ; col N = lane&15
    const int dcol = cj * 16 + m;
    const float sqd_d = sq[dcol];
#pragma unroll
    for (int r = 0; r < 8; ++r) {
      const int crow = ci * 16 + r + 8 * kh;
      const float g   = acc[r];
      float sqd = sq[crow] + sqd_d - 2.0f * g;
      sqd = fmaxf(sqd, 0.0f);
      const float dist = (sqd > 0.0f) ? sqrtf(sqd) : 0.0f;
      const float h = fmaxf(2.0f * D_DIST - dist, 0.0f);
      hinge = fmaf(h, h, hinge);
    }
  }

  // combine: P_VAR * var + P_DIST * hinge/denom + P_REG * lreg / L
  const float denom = 2.0f * (float)LBL * ((float)LBL - 1.0f);
  float part = P_DIST * hinge / denom
             + P_VAR * vterm / (float)LBL
             + P_REG * lreg_c / (float)LBL;

  red[tid] = part;
  __syncthreads();
  for (int s = 64; s > 0; s >>= 1) {
    if (tid < s) red[tid] += red[tid + s];
    __syncthreads();
  }
  if (tid == 0) {
    __hip_atomic_fetch_add(out, red[0], __ATOMIC_RELAXED, __HIP_MEMORY_SCOPE_AGENT);
  }
}

// ---------------------------------------------------------------------------
extern "C" void kernel_launch(void* const* d_in, const int* in_sizes, int n_in,
                              void* d_out, int out_size, void* d_ws, size_t ws_size,
                              hipStream_t stream) {
  (void)in_sizes; (void)n_in; (void)ws_size;
  const float* pred  = (const float*)d_in[0];   // [8,128,65536] f32
  const int*   label = (const int*)d_in[1];     // [8,65536] i32
  float* out = (float*)d_out;
  float* ws  = (float*)d_ws;

  // zero accumulators + output (must be done every call; harness poisons once)
  {
    const int n = WS_TOTAL;
    dl_zero_kernel<<<(n + 255) / 256, 256, 0, stream>>>(ws, out, n, out_size);
  }
  // streaming segmented-sum pass
  dl_pass1_kernel<<<BB * BLOCKS_PER_BATCH, T1, 0, stream>>>(pred, label, ws);
  // finalize with f32 WMMA Gram
  dl_finalize_kernel<<<BB, 128, 0, stream>>>(ws, out);
}